// CritiGraph_68951404970419
// MI455X (gfx1250) — compile-verified
//
#include <hip/hip_runtime.h>
#include <hip/hip_bf16.h>
#include <stdint.h>

#define HB     16
#define TPN    8
#define KKN    4
#define CCN    129     // 2*K*H+1
#define TTN    256
#define S_TOTN 128
#define S_COSN 96
#define CUR_TPN 4

typedef unsigned int v4u __attribute__((ext_vector_type(4)));
typedef int          v8i __attribute__((ext_vector_type(8)));
typedef int          v4i __attribute__((ext_vector_type(4)));

struct __align__(8) PA { uint32_t pos; float a; };   // one ds_load_b64 per (s,tp)

// frexp_exp(x+1) = 32 - clz(x+1)  =>  raw = 1 - exp/16 = (clz(x+1) - 16)/16
__device__ __forceinline__ float clzf_of(uint32_t x) {
    return (float)__clz((int)(x + 1u));
}

__global__ __launch_bounds__(256)
void critigraph_kernel(const long long*  __restrict__ sta_loc,
                       const long long*  __restrict__ pos_loc,
                       const float*      __restrict__ eu_val,
                       const float*      __restrict__ eu_norm,
                       const unsigned char* __restrict__ mask,
                       const long long*  __restrict__ rnd,
                       const float*      __restrict__ rand_vals,
                       const float*      __restrict__ t_rand,
                       float*            __restrict__ out)
{
    __shared__ uint32_t pos_l[S_TOTN * TPN];   // low dwords of pos_loc[t] (TDM-gathered)
    __shared__ uint32_t cnc[65 * TPN];         // result[0..63], sta at c=64 (negative half implicit)
    __shared__ float    cos_sp[S_TOTN * TPN];
    __shared__ PA       pa[S_TOTN * TPN];      // {pos, a}; s<96: a=(b-eu)*w ; s>=96: a=b
    __shared__ float2   cosS[S_COSN];          // {c8*w, -16*c8*w}, c8 = nrm8/16
    __shared__ float4   bceS[S_TOTN - S_COSN]; // {c8, -16*c8, w*eu, w*(1-eu)}
    __shared__ float    csp[S_TOTN];
    __shared__ float    euv[S_TOTN];
    __shared__ float    nrm8[S_TOTN];
    __shared__ float    nrmF[S_TOTN];
    __shared__ float    mskF[S_TOTN];
    __shared__ float    lossc[CCN * TPN];
    __shared__ float    lossr[CCN * TPN];
    __shared__ int      amin[TPN];
    __shared__ float    selC[TPN], selR[TPN];
    __shared__ float    invW[2];

    const int t   = blockIdx.x;
    const int tid = threadIdx.x;

    // ---------- Phase 0: TDM strided DMA: low dword of each int64 -> LDS ----------
#if __has_builtin(__builtin_amdgcn_tensor_load_to_lds) && __has_builtin(__builtin_amdgcn_s_wait_tensorcnt)
    if (tid < 32) {
        uint64_t ga = (uint64_t)(uintptr_t)(pos_loc + (size_t)t * S_TOTN * TPN);
        uint32_t lo = (uint32_t)(uintptr_t)&pos_l[0];
        v4u g0; v8i g1;
        v4i g2 = {0, 0, 0, 0};
        v4i g3 = {0, 0, 0, 0};
        g0[0] = 1u;                                                  // count=1, user D#
        g0[1] = lo;                                                  // lds_addr
        g0[2] = (uint32_t)ga;                                        // global_addr[31:0]
        g0[3] = (uint32_t)((ga >> 32) & 0x01FFFFFFu) | (2u << 30);   // addr[56:32], type=2
        g1[0] = (int)(2u << 16);      // data_size=4B; wg_mask=0 (not in cluster)
        g1[1] = (int)(2u << 16);      // tensor_dim0 = 2 dwords/row (bits 79:48)
        g1[2] = (int)(1024u << 16);   // tensor_dim1 = 1024 rows (bits 111:80)
        g1[3] = (int)(1u << 16);      // tile_dim0 = 1 dword (bits 127:112)
        g1[4] = (int)1024;            // tile_dim1 = 1024 rows; tile_dim2 = 0
        g1[5] = (int)2;               // tensor_dim0_stride = 2 dwords (8B)
        g1[6] = (int)(2u << 16);      // tensor_dim1_stride = 2 dwords (8B between rows)
        g1[7] = 0;
#if defined(__clang_major__) && (__clang_major__ >= 23)
        v8i g4 = {0, 0, 0, 0, 0, 0, 0, 0};
        __builtin_amdgcn_tensor_load_to_lds(g0, g1, g2, g3, g4, 0);
#else
        __builtin_amdgcn_tensor_load_to_lds(g0, g1, g2, g3, 0);
#endif
        __builtin_amdgcn_s_wait_tensorcnt((short)0);
    }
#else
    for (int i = tid; i < S_TOTN * TPN; i += 256)
        pos_l[i] = (uint32_t)(unsigned long long)pos_loc[(size_t)t * S_TOTN * TPN + i];
#endif

    // ---------- Phase 1: stage scalars + build connection table ----------
    if (tid < S_TOTN) {
        euv[tid]  = eu_val[(size_t)t * S_TOTN + tid];
        float nf  = eu_norm[(size_t)t * S_TOTN + tid];
        nrmF[tid] = nf;
        nrm8[tid] = nf * 0.125f;
        mskF[tid] = mask[(size_t)t * S_TOTN + tid] ? 1.0f : 0.0f;
    }
    if (tid < TPN)
        cnc[64 * TPN + tid] = (uint32_t)(unsigned long long)sta_loc[(size_t)t * TPN + tid];
    for (int i = tid; i < 64 * TPN; i += 256) {
        int c = i >> 3, tp = i & 7;
        int h = c >> 2, k = c & 3;
        uint32_t sta = (uint32_t)(unsigned long long)sta_loc[(size_t)t * TPN + tp];
        uint32_t rv  = (uint32_t)(unsigned long long)
            rnd[(((size_t)t * HB + h) * KKN + k) * TPN + tp];
        cnc[i] = sta ^ (1u << h) ^ (rv & ((1u << h) - 1u));
    }
    __syncthreads();   // TDM wave already did s_wait_tensorcnt 0 before arriving

    // ---------- Phase 2: cos_sta_pos, csp_sum, packed operand tables ----------
    for (int i = tid; i < S_TOTN * TPN; i += 256) {
        int s = i >> 3, tp = i & 7;
        float raw = (clzf_of(cnc[64 * TPN + tp] ^ pos_l[i]) - 16.0f) * 0.0625f;
        cos_sp[i] = raw * nrmF[s];
    }
    __syncthreads();
    if (tid < S_TOTN) {
        const float* r = &cos_sp[tid * TPN];
        float su = 0.f;
        #pragma unroll
        for (int j = 0; j < TPN; ++j) su += r[j];
        csp[tid] = su;
        // per-s coefficients for the main loop
        float c8 = nrm8[tid] * 0.0625f;      // d = (clz-16)*c8
        if (tid < S_COSN) {
            float cw = c8 * mskF[tid];
            cosS[tid] = make_float2(cw, -16.0f * cw);
        } else {
            float w = mskF[tid];
            bceS[tid - S_COSN] =
                make_float4(c8, -16.0f * c8, w * euv[tid], w * (1.0f - euv[tid]));
        }
    }
    if (tid == 0) {
        float lc = 0.f, lr = 0.f;
        for (int s = 0; s < S_COSN; ++s)      lc += mskF[s];
        for (int s = S_COSN; s < S_TOTN; ++s) lr += mskF[s];
        invW[0] = 1.0f / (lc + 1e-12f);
        invW[1] = 1.0f / (lr + 1e-12f);
    }
    __syncthreads();
    for (int i = tid; i < S_TOTN * TPN; i += 256) {
        int s = i >> 3;
        float b = (csp[s] - cos_sp[i]) * 0.125f;
        PA q;
        q.pos = pos_l[i];
        q.a   = (s < S_COSN) ? (b - euv[s]) * mskF[s] : b;
        pa[i] = q;
    }
    __syncthreads();

    // ---------- Phase 3: main loss loop; antisymmetric pair (c, c+65) ----------
    for (int it = tid; it < 65 * TPN; it += 256) {
        int c = it >> 3, tp = it & 7;
        uint32_t res = cnc[it];
        float negf = (res == 0u) ? 1.0f : -1.0f;   // dist(-res, pos) = negf * dist(res, pos)
        float lcp = 0.f, lcm = 0.f, lrp = 0.f, lrm = 0.f;
        #pragma unroll 4
        for (int s = 0; s < S_COSN; ++s) {
            PA     q  = pa[s * TPN + tp];
            float2 cs = cosS[s];
            float d   = fmaf(clzf_of(res ^ q.pos), cs.x, cs.y);
            float e1  = q.a + d;
            float e2  = fmaf(negf, d, q.a);
            lcp = fmaf(e1, e1, lcp);
            lcm = fmaf(e2, e2, lcm);
        }
        #pragma unroll 2
        for (int s = S_COSN; s < S_TOTN; ++s) {
            PA     q  = pa[s * TPN + tp];
            float4 bs = bceS[s - S_COSN];
            float d   = fmaf(clzf_of(res ^ q.pos), bs.x, bs.y);
            float ctp = q.a + d;
            float ctm = fmaf(negf, d, q.a);
            float pp = fminf(fmaxf(fmaf(ctp, 0.5f, 0.5f), 1e-6f), 1.0f - 1e-6f);
            float pm = fminf(fmaxf(fmaf(ctm, 0.5f, 0.5f), 1e-6f), 1.0f - 1e-6f);
            lrp -= bs.z * __logf(pp) + bs.w * log1pf(-pp);
            lrm -= bs.z * __logf(pm) + bs.w * log1pf(-pm);
        }
        lossc[it] = lcp * invW[0];
        lossr[it] = lrp * invW[1];
        if (c < 64) {
            lossc[(c + 65) * TPN + tp] = lcm * invW[0];
            lossr[(c + 65) * TPN + tp] = lrm * invW[1];
        }
    }
    __syncthreads();

    // ---------- Phase 4: argmin over c (first-min ties, ascending scan) ----------
    if (tid < TPN) {
        int tp = tid;
        float best = lossc[tp] + lossr[tp];
        int bi = 0;
        for (int c = 1; c < CCN; ++c) {
            float v = lossc[c * TPN + tp] + lossr[c * TPN + tp];
            if (v < best) { best = v; bi = c; }
        }
        amin[tp] = bi;
    }
    __syncthreads();

    // ---------- Phase 5: stable top-4 rank selection + output gather ----------
    if (tid < TPN) {
        int tp = tid;
        float rv = rand_vals[(size_t)t * TPN + tp];
        int rank = 0;
        for (int j = 0; j < TPN; ++j) {
            float rj = rand_vals[(size_t)t * TPN + j];
            rank += (rj < rv) || (rj == rv && j < tp);
        }
        bool sel = (rank < CUR_TPN) && (t_rand[t] < 0.5f);
        int idx = sel ? amin[tp] : 64;
        float loc = (idx <= 64) ? (float)cnc[idx * TPN + tp]
                                : -(float)cnc[(idx - 65) * TPN + tp];
        out[(size_t)t * TPN + tp] = loc;
        selC[tp] = lossc[idx * TPN + tp];
        selR[tp] = lossr[idx * TPN + tp];
    }
    __syncthreads();
    if (tid == 0) {
        float sc = 0.f, sr = 0.f;
        #pragma unroll
        for (int j = 0; j < TPN; ++j) { sc += selC[j]; sr += selR[j]; }
        sc *= (1.0f / TPN);
        sr *= (1.0f / TPN);
        out[TTN * TPN + t]           = sc;
        out[TTN * TPN + TTN + t]     = sr;
        out[TTN * TPN + 2 * TTN + t] = sc + sr;
    }
}

extern "C" void kernel_launch(void* const* d_in, const int* in_sizes, int n_in,
                              void* d_out, int out_size, void* d_ws, size_t ws_size,
                              hipStream_t stream) {
    (void)in_sizes; (void)n_in; (void)out_size; (void)d_ws; (void)ws_size;
    const long long*     sta = (const long long*)d_in[0];
    const long long*     pos = (const long long*)d_in[1];
    const float*         euv = (const float*)d_in[2];
    const float*         eun = (const float*)d_in[3];
    const unsigned char* msk = (const unsigned char*)d_in[4];
    const long long*     rnd = (const long long*)d_in[5];
    const float*         rvl = (const float*)d_in[6];
    const float*         trd = (const float*)d_in[7];
    float* out = (float*)d_out;
    critigraph_kernel<<<TTN, 256, 0, stream>>>(sta, pos, euv, eun, msk, rnd, rvl, trd, out);
}